// GradCAMPP_8022998909038
// MI455X (gfx1250) — compile-verified
//
#include <hip/hip_runtime.h>
#include <hip/hip_bf16.h>

#define EPSV 1e-6f
#define THRV 0.35f

typedef __attribute__((ext_vector_type(2))) float v2f;
typedef __attribute__((ext_vector_type(8))) float v8f;

#define NB   16
#define CIN  32
#define HIN  224
#define WIN  224
#define HWIN (HIN*WIN)    // 50176
#define CF   128
#define HO   112
#define WO   112
#define HWO  (HO*WO)      // 12544
#define NCLS 1000
#define KDIM 288          // CIN*9
#define FLT_BIG 3.402823466e+38f

#define TILE_COLS 34
#define LDS_ELEMS (CIN * 3 * TILE_COLS)   // 3264 floats = 13 KB

// ---------------------------------------------------------------------------
// Conv 3x3 stride-2 SAME as implicit GEMM via V_WMMA_F32_16X16X4_F32.
// One 128-thread block (4 waves) handles one (oh, ow-tile, n): the shared
// 32ci x 3row x 34col input window is staged into LDS once (zero-filled for
// SAME padding), then each wave computes a 32(cf) x 16(ow) output tile with
// two accumulators. K-reduction is tap-major (k' = tap*32 + ci) so the inner
// unrolled loop is pure ds_load + weight load + 2x WMMA, no conditionals.
// A-frag: lanes 0-15 = M rows, lane>>4 picks the K pair {0,1} vs {2,3}.
// B-frag: lanes 0-15 = N cols, same K-pair split.
// ---------------------------------------------------------------------------
__global__ __launch_bounds__(128) void conv_wmma_k(
    const float* __restrict__ x, const float* __restrict__ Wc,
    const float* __restrict__ bc, float* __restrict__ A)
{
  const int tid   = threadIdx.x;
  const int lane  = tid & 31;
  const int wid   = tid >> 5;              // cf-tile group 0..3 (32 cf each)
  const int tile  = blockIdx.x;            // 0..783 : oh*7 + ow_tile
  const int n     = blockIdx.y;            // 0..15
  const int oh    = tile / 7;
  const int ow0   = (tile - oh * 7) * 16;
  const int m     = lane & 15;             // M row (A) == N col (B/C/D)
  const int khalf = lane >> 4;             // selects K pair

  __shared__ float sx[LDS_ELEMS];

  const float* __restrict__ xn = x + (size_t)n * CIN * HWIN;

  // Cooperative stage: x[ci, 2*oh+row, 2*ow0+col] -> sx[(ci*3+row)*34 + col]
  for (int e = tid; e < LDS_ELEMS; e += 128) {
    int ci  = e / (3 * TILE_COLS);
    int rem = e - ci * (3 * TILE_COLS);
    int row = rem / TILE_COLS;
    int col = rem - row * TILE_COLS;
    int ih = 2 * oh + row, iw = 2 * ow0 + col;   // SAME: pad_low = 0
    sx[e] = (ih < HIN && iw < WIN)
              ? xn[(size_t)ci * HWIN + (size_t)ih * WIN + iw] : 0.f;
  }

  const float* __restrict__ w0 = Wc + (size_t)(wid * 32 + m) * KDIM;       // rows 0..15
  const float* __restrict__ w1 = Wc + (size_t)(wid * 32 + 16 + m) * KDIM;  // rows 16..31
  __builtin_prefetch(w0, 0, 3);   // global_prefetch_b8 (weights are L2-hot)
  __builtin_prefetch(w1, 0, 3);

  __syncthreads();

  v8f acc0 = {0.f, 0.f, 0.f, 0.f, 0.f, 0.f, 0.f, 0.f};
  v8f acc1 = {0.f, 0.f, 0.f, 0.f, 0.f, 0.f, 0.f, 0.f};

#pragma unroll
  for (int r = 0; r < 9; ++r) {            // tap index: kh*3+kw
    const int kh = r / 3, kw = r - kh * 3;
    const int colb = 2 * m + kw;           // this lane's column in the window
#pragma unroll
    for (int s = 0; s < 8; ++s) {          // ci steps of 4
      const int ci0 = 4 * s + 2 * khalf;   // this lane's K pair within step
      v2f a0, a1, b;
      a0.x = w0[ci0 * 9 + r];
      a0.y = w0[(ci0 + 1) * 9 + r];
      a1.x = w1[ci0 * 9 + r];
      a1.y = w1[(ci0 + 1) * 9 + r];
      b.x = sx[(ci0 * 3 + kh) * TILE_COLS + colb];
      b.y = sx[((ci0 + 1) * 3 + kh) * TILE_COLS + colb];
      acc0 = __builtin_amdgcn_wmma_f32_16x16x4_f32(false, a0, false, b,
                                                   (short)0, acc0, false, false);
      acc1 = __builtin_amdgcn_wmma_f32_16x16x4_f32(false, a1, false, b,
                                                   (short)0, acc1, false, false);
    }
  }

  // C/D layout: lane l, vgpr v -> M = v + 8*(l>>4), N = l&15
  const int ow = ow0 + m;
  const size_t pix = (size_t)oh * WO + ow;
#pragma unroll
  for (int v = 0; v < 8; ++v) {
    int cfa = wid * 32 + 8 * khalf + v;
    int cfb = cfa + 16;
    A[((size_t)n * CF + cfa) * HWO + pix] = acc0[v] + bc[cfa];
    A[((size_t)n * CF + cfb) * HWO + pix] = acc1[v] + bc[cfb];
  }
}

// ---- sumA[n,c] = sum_hw A --------------------------------------------------
__global__ void sum_spatial_k(const float* __restrict__ A, float* __restrict__ sumA) {
  int nc = blockIdx.x;
  const float* p = A + (size_t)nc * HWO;
  float s = 0.f;
  for (int i = threadIdx.x; i < HWO; i += 256) s += p[i];
  __shared__ float sm[256];
  sm[threadIdx.x] = s; __syncthreads();
  for (int off = 128; off > 0; off >>= 1) {
    if ((int)threadIdx.x < off) sm[threadIdx.x] += sm[threadIdx.x + off];
    __syncthreads();
  }
  if (threadIdx.x == 0) sumA[nc] = sm[0];
}

// ---- logits[n,cls] = mean(A) . Wf[cls] + bf --------------------------------
__global__ void logits_k(const float* __restrict__ sumA, const float* __restrict__ Wf,
                         const float* __restrict__ bf, float* __restrict__ logits) {
  int idx = blockIdx.x * blockDim.x + threadIdx.x;
  if (idx >= NB * NCLS) return;
  int n = idx / NCLS, cls = idx - n * NCLS;
  const float invHW = 1.f / (float)HWO;
  float s = bf[cls];
  for (int c = 0; c < CF; ++c)
    s += (sumA[n * CF + c] * invHW) * Wf[cls * CF + c];
  logits[idx] = s;
}

// ---- top-2 per sample (ties -> lowest index, like lax.top_k) ---------------
__global__ void top2_k(const float* __restrict__ logits, int* __restrict__ lab,
                       float* __restrict__ labf) {
  int n = blockIdx.x, tid = threadIdx.x;
  const float* lg = logits + n * NCLS;
  __shared__ float sv[256]; __shared__ int si[256]; __shared__ int s_i0;

  float bv = -FLT_BIG; int bi = NCLS;
  for (int i = tid; i < NCLS; i += 256) {
    float v = lg[i];
    if (v > bv || (v == bv && i < bi)) { bv = v; bi = i; }
  }
  sv[tid] = bv; si[tid] = bi; __syncthreads();
  for (int off = 128; off > 0; off >>= 1) {
    if (tid < off && (sv[tid + off] > sv[tid] ||
        (sv[tid + off] == sv[tid] && si[tid + off] < si[tid]))) {
      sv[tid] = sv[tid + off]; si[tid] = si[tid + off];
    }
    __syncthreads();
  }
  if (tid == 0) { s_i0 = si[0]; lab[n * 2] = si[0]; labf[n * 2] = (float)si[0]; }
  __syncthreads();
  int i0 = s_i0;

  bv = -FLT_BIG; bi = NCLS;
  for (int i = tid; i < NCLS; i += 256) {
    if (i == i0) continue;
    float v = lg[i];
    if (v > bv || (v == bv && i < bi)) { bv = v; bi = i; }
  }
  sv[tid] = bv; si[tid] = bi; __syncthreads();
  for (int off = 128; off > 0; off >>= 1) {
    if (tid < off && (sv[tid + off] > sv[tid] ||
        (sv[tid + off] == sv[tid] && si[tid + off] < si[tid]))) {
      sv[tid] = sv[tid + off]; si[tid] = si[tid + off];
    }
    __syncthreads();
  }
  if (tid == 0) { lab[n * 2 + 1] = si[0]; labf[n * 2 + 1] = (float)si[0]; }
}

// ---- Grad-CAM++ channel weights: G is spatially constant -------------------
__global__ void wv_k(const float* __restrict__ Wf, const int* __restrict__ lab,
                     const float* __restrict__ sumA, float* __restrict__ wv) {
  int idx = blockIdx.x * blockDim.x + threadIdx.x;  // k*2048 + n*128 + c
  if (idx >= 2 * NB * CF) return;
  int k = idx >> 11;
  int r = idx & 2047;
  int n = r >> 7, c = r & 127;
  const float invHW = 1.f / (float)HWO;
  float G  = Wf[lab[n * 2 + k] * CF + c] * invHW;
  float g2 = G * G, g3 = g2 * G;
  float denom = 2.f * g2 + g3 * sumA[n * CF + c] + EPSV;
  float alpha = g2 / denom;
  float rg = G > 0.f ? G : 0.f;
  wv[idx] = alpha * rg * (float)HWO;   // w = sum_hw(alpha*relu(G))
}

// ---- cam[k,n,p] = relu(sum_c w * A), both k in one pass over A -------------
__global__ void cam_k(const float* __restrict__ A, const float* __restrict__ wv,
                      float* __restrict__ cam) {
  int p = blockIdx.x * blockDim.x + threadIdx.x;
  if (p >= HWO) return;
  int n = blockIdx.y;
  const float* w0 = wv + (size_t)(0 * NB + n) * CF;
  const float* w1 = wv + (size_t)(1 * NB + n) * CF;
  const float* An = A + (size_t)n * CF * HWO + p;
  float s0 = 0.f, s1 = 0.f;
  for (int c = 0; c < CF; ++c) {
    float av = An[(size_t)c * HWO];
    s0 += w0[c] * av;
    s1 += w1[c] * av;
  }
  cam[(size_t)(0 * NB + n) * HWO + p] = s0 > 0.f ? s0 : 0.f;
  cam[(size_t)(1 * NB + n) * HWO + p] = s1 > 0.f ? s1 : 0.f;
}

// ---- per-map min/max -------------------------------------------------------
__global__ void minmax_k(const float* __restrict__ cam, float* __restrict__ mnmx) {
  int b = blockIdx.x;  // k*NB + n
  const float* cm = cam + (size_t)b * HWO;
  float mn = FLT_BIG, mx = -FLT_BIG;
  for (int i = threadIdx.x; i < HWO; i += 256) {
    float v = cm[i]; mn = fminf(mn, v); mx = fmaxf(mx, v);
  }
  __shared__ float smn[256], smx[256];
  smn[threadIdx.x] = mn; smx[threadIdx.x] = mx; __syncthreads();
  for (int off = 128; off > 0; off >>= 1) {
    if ((int)threadIdx.x < off) {
      smn[threadIdx.x] = fminf(smn[threadIdx.x], smn[threadIdx.x + off]);
      smx[threadIdx.x] = fmaxf(smx[threadIdx.x], smx[threadIdx.x + off]);
    }
    __syncthreads();
  }
  if (threadIdx.x == 0) { mnmx[b * 2] = smn[0]; mnmx[b * 2 + 1] = smx[0]; }
}

// ---- fused bilinear-upsample (half-pixel) + threshold + bbox + score -------
__global__ void bbox_k(const float* __restrict__ cam, const float* __restrict__ mnmx,
                       float* __restrict__ boxes, float* __restrict__ scores,
                       float* __restrict__ validf) {
  int b = blockIdx.x;                 // n*2 + k
  int n = b >> 1, k = b & 1;
  int tid = threadIdx.x;
  const float* cm = cam + (size_t)(k * NB + n) * HWO;
  float mn = mnmx[(k * NB + n) * 2], mx = mnmx[(k * NB + n) * 2 + 1];
  float inv = 1.f / (mx - mn + EPSV);

  float x1 = FLT_BIG, y1 = FLT_BIG, x2 = -FLT_BIG, y2 = -FLT_BIG, sum = 0.f;
  int cnt = 0;
  for (int p = tid; p < HIN * WIN; p += 256) {
    int i = p / WIN, j = p - i * WIN;
    float sy = i * 0.5f - 0.25f, sx = j * 0.5f - 0.25f;
    int y0 = (int)floorf(sy), x0 = (int)floorf(sx);
    float fy = sy - (float)y0, fx = sx - (float)x0;
    int y0c = min(max(y0, 0), HO - 1), y1c = min(max(y0 + 1, 0), HO - 1);
    int x0c = min(max(x0, 0), WO - 1), x1c = min(max(x0 + 1, 0), WO - 1);
    float v00 = cm[y0c * WO + x0c], v01 = cm[y0c * WO + x1c];
    float v10 = cm[y1c * WO + x0c], v11 = cm[y1c * WO + x1c];
    float v = v00 * (1.f - fy) * (1.f - fx) + v01 * (1.f - fy) * fx
            + v10 * fy * (1.f - fx)         + v11 * fy * fx;
    v = (v - mn) * inv;
    if (v >= THRV) {
      x1 = fminf(x1, (float)j); y1 = fminf(y1, (float)i);
      x2 = fmaxf(x2, (float)j); y2 = fmaxf(y2, (float)i);
      sum += v; cnt++;
    }
  }
  __shared__ float sx1[256], sy1[256], sx2[256], sy2[256], ssum[256];
  __shared__ int scnt[256];
  sx1[tid] = x1; sy1[tid] = y1; sx2[tid] = x2; sy2[tid] = y2;
  ssum[tid] = sum; scnt[tid] = cnt;
  __syncthreads();
  for (int off = 128; off > 0; off >>= 1) {
    if (tid < off) {
      sx1[tid] = fminf(sx1[tid], sx1[tid + off]);
      sy1[tid] = fminf(sy1[tid], sy1[tid + off]);
      sx2[tid] = fmaxf(sx2[tid], sx2[tid + off]);
      sy2[tid] = fmaxf(sy2[tid], sy2[tid + off]);
      ssum[tid] += ssum[tid + off];
      scnt[tid] += scnt[tid + off];
    }
    __syncthreads();
  }
  if (tid == 0) {
    bool valid = scnt[0] > 0;
    float* bx = boxes + b * 4;
    bx[0] = valid ? sx1[0] : -1.f;
    bx[1] = valid ? sy1[0] : -1.f;
    bx[2] = valid ? sx2[0] : -1.f;
    bx[3] = valid ? sy2[0] : -1.f;
    scores[b] = valid ? ssum[0] / (float)max(scnt[0], 1) : 0.f;
    validf[b] = valid ? 1.f : 0.f;
  }
}

extern "C" void kernel_launch(void* const* d_in, const int* in_sizes, int n_in,
                              void* d_out, int out_size, void* d_ws, size_t ws_size,
                              hipStream_t stream) {
  const float* x  = (const float*)d_in[0];
  const float* Wc = (const float*)d_in[1];
  const float* bc = (const float*)d_in[2];
  const float* Wf = (const float*)d_in[3];
  const float* bf = (const float*)d_in[4];
  // d_in[5] = topk (always 2 for this reference)

  float* out     = (float*)d_out;
  float* logits  = out;               // 16*1000
  float* boxes   = out + 16000;       // 16*2*4
  float* labelsf = out + 16128;       // 16*2
  float* scores  = out + 16160;       // 16*2
  float* validf  = out + 16192;       // 16*2

  float* ws   = (float*)d_ws;
  float* A    = ws;                                   // 16*128*12544
  float* cam  = A    + (size_t)NB * CF * HWO;         // 2*16*12544
  float* sumA = cam  + (size_t)2 * NB * HWO;          // 16*128
  float* wv   = sumA + (size_t)NB * CF;               // 2*16*128
  float* mnmx = wv   + (size_t)2 * NB * CF;           // 2*16*2
  int*   lab  = (int*)(mnmx + (size_t)2 * NB * 2);    // 16*2

  conv_wmma_k<<<dim3(784, NB), 128, 0, stream>>>(x, Wc, bc, A);
  sum_spatial_k<<<NB * CF, 256, 0, stream>>>(A, sumA);
  logits_k<<<(NB * NCLS + 255) / 256, 256, 0, stream>>>(sumA, Wf, bf, logits);
  top2_k<<<NB, 256, 0, stream>>>(logits, lab, labelsf);
  wv_k<<<(2 * NB * CF + 255) / 256, 256, 0, stream>>>(Wf, lab, sumA, wv);
  cam_k<<<dim3((HWO + 255) / 256, NB), 256, 0, stream>>>(A, wv, cam);
  minmax_k<<<2 * NB, 256, 0, stream>>>(cam, mnmx);
  bbox_k<<<2 * NB, 256, 0, stream>>>(cam, mnmx, boxes, scores, validf);
}